// GAT_28518582845442
// MI455X (gfx1250) — compile-verified
//
#include <hip/hip_runtime.h>
#include <hip/hip_bf16.h>
#include <cfloat>

typedef __attribute__((ext_vector_type(2))) float v2f;
typedef __attribute__((ext_vector_type(8))) float v8f;

#define NNODES 50000
#define NHEADS 3
#define NEG_SLOPE 0.2f
#define TN 4   // 16-col tiles per wave

// ---------------------------------------------------------------- utilities

__global__ void gat_fill_f32(float* __restrict__ p, float v, int n) {
  int i = blockIdx.x * blockDim.x + threadIdx.x;
  if (i < n) p[i] = v;
}

__device__ __forceinline__ void atomicMaxF32(float* addr, float val) {
  // IEEE bit-pattern ordering trick; buffer initialized to -FLT_MAX.
  if (val >= 0.0f) {
    atomicMax((int*)addr, __float_as_int(val));
  } else {
    atomicMin((unsigned int*)addr, (unsigned int)__float_as_int(val));
  }
}

// Pack W[K,Nout] into WMMA-B fragment order (zero-padded to Ntiles*16 cols):
//   Bf[ ((t*(K/4)+kg)*32 + lane)*2 + pos ] = W[kg*4 + (lane>=16)*2 + pos, t*16 + lane%16]
// so each wave B-fragment load is one contiguous 256B burst and the per-lane
// k-step stride is a constant +64 floats.
__global__ void gat_pack_bfrag(const float* __restrict__ W,
                               float* __restrict__ Bf,
                               int K, int Nout, int Ntiles) {
  int total = Ntiles * (K >> 2) * 64;
  int i = blockIdx.x * blockDim.x + threadIdx.x;
  if (i >= total) return;
  int pos  = i & 1;
  int lane = (i >> 1) & 31;
  int kg   = (i >> 6) % (K >> 2);
  int t    = (i >> 6) / (K >> 2);
  int c = t * 16 + (lane & 15);
  int k = kg * 4 + ((lane >> 4) << 1) + pos;
  Bf[i] = (c < Nout) ? W[(size_t)k * Nout + c] : 0.0f;
}

// ---------------------------------------------------------------- fp32 WMMA GEMM
// C[M,Nout] = A[M,K] * W. One wave computes a 16 x (16*TN) strip with TN
// independent accumulators. Software-pipelined: next k-step's A + 4 B frags
// are prefetched into separate registers while the current 4 WMMAs execute.
__global__ void gat_gemm_f32_wmma(const float* __restrict__ A,
                                  const float* __restrict__ Bf,  // packed frags
                                  float* __restrict__ C,
                                  int M, int K, int Nout,
                                  int Ngroups, int totalWaves) {
  int wave = (blockIdx.x * blockDim.x + threadIdx.x) >> 5;
  if (wave >= totalWaves) return;            // whole-wave uniform exit
  int lane = threadIdx.x & 31;
  int l15  = lane & 15;
  int koff = (lane >> 4) << 1;               // lanes 16..31 hold K+2,K+3
  int mt = wave / Ngroups;
  int ng = wave - mt * Ngroups;
  int row = mt * 16 + l15;                   // A row for this lane
  int t0  = ng * TN;                         // first column tile index
  int Kg  = K >> 2;

  const float* __restrict__ ap = A + (size_t)row * K + koff;
  const float* bp[TN];
#pragma unroll
  for (int j = 0; j < TN; ++j)
    bp[j] = Bf + (size_t)(t0 + j) * Kg * 64 + lane * 2;

  v8f acc[TN] = {};

  // pipeline prologue
  v2f a0 = *reinterpret_cast<const v2f*>(ap);
  v2f b0[TN];
#pragma unroll
  for (int j = 0; j < TN; ++j) b0[j] = *reinterpret_cast<const v2f*>(bp[j]);

  for (int it = 0; it < Kg - 1; ++it) {
    // prefetch next k-step (in-bounds: last prefetch reads kg = Kg-1)
    v2f a1 = *reinterpret_cast<const v2f*>(ap + 4);
    v2f b1[TN];
#pragma unroll
    for (int j = 0; j < TN; ++j)
      b1[j] = *reinterpret_cast<const v2f*>(bp[j] + 64);

#pragma unroll
    for (int j = 0; j < TN; ++j)
      acc[j] = __builtin_amdgcn_wmma_f32_16x16x4_f32(
          false, a0, false, b0[j], (short)0, acc[j], false, false);

    a0 = a1;
#pragma unroll
    for (int j = 0; j < TN; ++j) b0[j] = b1[j];
    ap += 4;
#pragma unroll
    for (int j = 0; j < TN; ++j) bp[j] += 64;
  }
  // epilogue
#pragma unroll
  for (int j = 0; j < TN; ++j)
    acc[j] = __builtin_amdgcn_wmma_f32_16x16x4_f32(
        false, a0, false, b0[j], (short)0, acc[j], false, false);

  int srow = mt * 16 + ((lane >> 4) << 3);   // acc[j][r] -> rows {r, r+8}
#pragma unroll
  for (int j = 0; j < TN; ++j) {
    int col = (t0 + j) * 16 + l15;
    if (col < Nout) {                        // layer 2 pad tiles skipped
#pragma unroll
      for (int r = 0; r < 8; ++r)
        C[(size_t)(srow + r) * Nout + col] = acc[j][r];
    }
  }
}

// ---------------------------------------------------------------- attention
__global__ void gat_attn_coef(const float* __restrict__ feat,
                              const float* __restrict__ al,
                              const float* __restrict__ ar,
                              float* __restrict__ el, float* __restrict__ er,
                              int NH, int F, int H) {
  int i = blockIdx.x * blockDim.x + threadIdx.x;
  if (i >= NH) return;
  int h = i % H;
  const float* fp  = feat + (size_t)i * F;   // feat is [N,H,F] contiguous
  const float* alp = al + h * F;
  const float* arp = ar + h * F;
  float sl = 0.0f, sr = 0.0f;
  for (int f = 0; f < F; ++f) {
    float v = fp[f];
    sl += v * alp[f];
    sr += v * arp[f];
  }
  el[i] = sl;
  er[i] = sr;
}

// e = leaky_relu(el[src]+er[dst]); segment max into emax[dst]
__global__ void gat_edge_logits(const int* __restrict__ src,
                                const int* __restrict__ dst,
                                const float* __restrict__ el,
                                const float* __restrict__ er,
                                float* __restrict__ ebuf,
                                float* __restrict__ emax,
                                int EH, int H) {
  int i = blockIdx.x * blockDim.x + threadIdx.x;
  if (i >= EH) return;
  int e = i / H, h = i - e * H;
  int s = src[e], d = dst[e];
  float v = el[s * H + h] + er[d * H + h];
  v = (v > 0.0f) ? v : NEG_SLOPE * v;
  ebuf[i] = v;
  atomicMaxF32(&emax[d * H + h], v);
}

// ex = exp(e - emax[dst]); segment sum into denom[dst]
__global__ void gat_edge_exp(const int* __restrict__ dst,
                             const float* __restrict__ emax,
                             float* __restrict__ ebuf,
                             float* __restrict__ denom,
                             int EH, int H) {
  int i = blockIdx.x * blockDim.x + threadIdx.x;
  if (i >= EH) return;
  int e = i / H, h = i - e * H;
  int d = dst[e];
  float ex = __expf(ebuf[i] - emax[d * H + h]);
  ebuf[i] = ex;
  atomicAdd(&denom[d * H + h], ex);
}

// alpha = ex / denom[dst]
__global__ void gat_edge_alpha(const int* __restrict__ dst,
                               const float* __restrict__ denom,
                               float* __restrict__ ebuf,
                               int EH, int H) {
  int i = blockIdx.x * blockDim.x + threadIdx.x;
  if (i >= EH) return;
  int e = i / H, h = i - e * H;
  ebuf[i] = ebuf[i] / denom[dst[e] * H + h];
}

// out[dst, h, 4f..4f+3] += feat[src, h, 4f..4f+3] * alpha[e, h]
// float4 gathers (rows are 16B aligned: HF in {192,120}); scalar f32 atomics.
__global__ void gat_edge_aggregate4(const int* __restrict__ src,
                                    const int* __restrict__ dst,
                                    const float* __restrict__ feat,
                                    const float* __restrict__ alpha,
                                    float* __restrict__ out,
                                    long long total4, int HF4, int Fq, int H) {
  long long idx = (long long)blockIdx.x * blockDim.x + threadIdx.x;
  if (idx >= total4) return;
  int e  = (int)(idx / HF4);
  int r4 = (int)(idx - (long long)e * HF4);
  int h  = r4 / Fq;
  float a = alpha[(size_t)e * H + h];
  const float4* fp = (const float4*)(feat + (size_t)src[e] * (HF4 * 4)) + r4;
  float4 v = *fp;
  float* op = out + (size_t)dst[e] * (HF4 * 4) + (size_t)r4 * 4;
  atomicAdd(op + 0, v.x * a);
  atomicAdd(op + 1, v.y * a);
  atomicAdd(op + 2, v.z * a);
  atomicAdd(op + 3, v.w * a);
}

__global__ void gat_relu(float* __restrict__ p, int n) {
  int i = blockIdx.x * blockDim.x + threadIdx.x;
  if (i < n) p[i] = fmaxf(p[i], 0.0f);
}

// logits[n,c] = mean over heads of acc[n,h,c]
__global__ void gat_mean_heads(const float* __restrict__ acc,
                               float* __restrict__ out, int N, int C) {
  int i = blockIdx.x * blockDim.x + threadIdx.x;
  if (i >= N * C) return;
  int n = i / C, c = i - n * C;
  const float* p = acc + (size_t)n * NHEADS * C;
  out[i] = (p[c] + p[C + c] + p[2 * C + c]) * (1.0f / 3.0f);
}

// ---------------------------------------------------------------- host side

static inline int cdiv(long long a, long long b) { return (int)((a + b - 1) / b); }

static void run_gat_layer(const float* h, int K,
                          const float* W, const float* al, const float* ar, int F,
                          const int* src, const int* dst, int E,
                          float* feat, float* Bf, float* el, float* er,
                          float* emax, float* denom, float* ebuf,
                          float* out, bool relu_act, hipStream_t stream) {
  const int H = NHEADS, N = NNODES;
  const int HF = H * F;
  const int Ntiles  = cdiv(HF, 16);              // 12 or 8
  const int Ngroups = Ntiles / TN;               // 3 or 2

  // 0) pack weights into WMMA-B fragment order (zero-padded)
  gat_pack_bfrag<<<cdiv((long long)Ntiles * (K / 4) * 64, 256), 256, 0, stream>>>(
      W, Bf, K, HF, Ntiles);

  // 1) feat = h @ W  (fp32 WMMA, TN column tiles per wave, pipelined)
  int Mtiles = N / 16;                           // 3125 exactly
  int totalWaves = Mtiles * Ngroups;
  gat_gemm_f32_wmma<<<cdiv((long long)totalWaves * 32, 128), 128, 0, stream>>>(
      h, Bf, feat, N, K, HF, Ngroups, totalWaves);

  // 2) init per-node accumulators
  gat_fill_f32<<<cdiv(N * H, 256), 256, 0, stream>>>(emax, -FLT_MAX, N * H);
  gat_fill_f32<<<cdiv(N * H, 256), 256, 0, stream>>>(denom, 0.0f, N * H);
  gat_fill_f32<<<cdiv((long long)N * HF, 256), 256, 0, stream>>>(out, 0.0f, N * HF);

  // 3) attention coefficients
  gat_attn_coef<<<cdiv(N * H, 256), 256, 0, stream>>>(feat, al, ar, el, er, N * H, F, H);

  // 4) segment softmax over edges
  gat_edge_logits<<<cdiv((long long)E * H, 256), 256, 0, stream>>>(src, dst, el, er, ebuf, emax, E * H, H);
  gat_edge_exp   <<<cdiv((long long)E * H, 256), 256, 0, stream>>>(dst, emax, ebuf, denom, E * H, H);
  gat_edge_alpha <<<cdiv((long long)E * H, 256), 256, 0, stream>>>(dst, denom, ebuf, E * H, H);

  // 5) weighted aggregation (dominant phase; feat/out live in L2)
  long long total4 = (long long)E * (HF / 4);
  gat_edge_aggregate4<<<cdiv(total4, 256), 256, 0, stream>>>(
      src, dst, feat, ebuf, out, total4, HF / 4, F / 4, H);

  if (relu_act)
    gat_relu<<<cdiv((long long)N * HF, 256), 256, 0, stream>>>(out, N * HF);
}

extern "C" void kernel_launch(void* const* d_in, const int* in_sizes, int n_in,
                              void* d_out, int out_size, void* d_ws, size_t ws_size,
                              hipStream_t stream) {
  const float* x   = (const float*)d_in[0];   // [N, 256]
  const int*   src = (const int*)  d_in[1];   // [E]
  const int*   dst = (const int*)  d_in[2];   // [E]
  const float* W0  = (const float*)d_in[3];   // [256, 192]
  const float* al0 = (const float*)d_in[4];   // [3, 64]
  const float* ar0 = (const float*)d_in[5];
  const float* W1  = (const float*)d_in[6];   // [192, 192]
  const float* al1 = (const float*)d_in[7];
  const float* ar1 = (const float*)d_in[8];
  const float* W2  = (const float*)d_in[9];   // [192, 120]
  const float* al2 = (const float*)d_in[10];  // [3, 40]
  const float* ar2 = (const float*)d_in[11];

  const int N = NNODES, H = NHEADS;
  const int E = in_sizes[1];

  // workspace carve-up (floats)
  float* ws    = (float*)d_ws;
  float* feat  = ws;                              // N*192
  float* bufA  = feat  + (size_t)N * 192;         // N*192  (layer0 out / layer2 out)
  float* bufB  = bufA  + (size_t)N * 192;         // N*192  (layer1 out)
  float* el    = bufB  + (size_t)N * 192;         // N*3
  float* er    = el    + (size_t)N * H;           // N*3
  float* emax  = er    + (size_t)N * H;           // N*3
  float* denom = emax  + (size_t)N * H;           // N*3
  float* ebuf  = denom + (size_t)N * H;           // E*3
  float* Bf    = ebuf  + (size_t)E * H;           // 192*256 (packed W frags)

  // layer 0: [N,256] -> [N,3,64], relu
  run_gat_layer(x,    256, W0, al0, ar0, 64, src, dst, E,
                feat, Bf, el, er, emax, denom, ebuf, bufA, true, stream);
  // layer 1: [N,192] -> [N,3,64], relu
  run_gat_layer(bufA, 192, W1, al1, ar1, 64, src, dst, E,
                feat, Bf, el, er, emax, denom, ebuf, bufB, true, stream);
  // layer 2: [N,192] -> [N,3,40], no activation
  run_gat_layer(bufB, 192, W2, al2, ar2, 40, src, dst, E,
                feat, Bf, el, er, emax, denom, ebuf, bufA, false, stream);

  // logits = mean over heads -> [N, 40]
  gat_mean_heads<<<cdiv(N * 40, 256), 256, 0, stream>>>(bufA, (float*)d_out, N, 40);
}